// DecodeGPT2Attention_25735444038105
// MI455X (gfx1250) — compile-verified
//
#include <hip/hip_runtime.h>
#include <hip/hip_bf16.h>
#include <math.h>

// ---------------------------------------------------------------------------
// DecodeGPT2Attention for MI455X (gfx1250, wave32)
// B=16, E=2048, H=16, D=128, S_old=4095, S=4096
// Outputs (concatenated in d_out): out[16*2048], keys[16*4096*2048],
//                                  values[16*4096*2048]
// Memory-bound workload (~3.3 GB traffic @ 23.3 TB/s); GEMMs use native fp32
// WMMA (v_wmma_f32_16x16x4_f32) since M=16 == one tile and precision is free.
// ---------------------------------------------------------------------------

typedef __attribute__((ext_vector_type(2))) float v2f;
typedef __attribute__((ext_vector_type(8))) float v8f;

#define E_DIM   2048
#define S_DIM   4096
#define S_OLD   4095
#define B_DIM   16
#define H_DIM   16
#define D_DIM   128
#define QKV_N   6144

// --------------------------------------------------------------------------
// Kernel 0: last[b] = sum(attention_mask[b,:]) - 1
// --------------------------------------------------------------------------
__global__ void last_kernel(const int* __restrict__ mask, int* __restrict__ last) {
    __shared__ int red[256];
    const int b = blockIdx.x;
    const int tid = threadIdx.x;
    int s = 0;
    for (int i = tid; i < S_DIM; i += 256) s += mask[(size_t)b * S_DIM + i];
    red[tid] = s;
    __syncthreads();
    for (int w = 128; w >= 1; w >>= 1) {
        if (tid < w) red[tid] += red[tid + w];
        __syncthreads();
    }
    if (tid == 0) last[b] = red[0] - 1;
}

// --------------------------------------------------------------------------
// Kernel 1/5: C(16 x N) = A(16 x K) * W(K x N) + bias, via f32 WMMA 16x16x4.
// One wave per 16-wide N tile.  ISA layouts:
//   A 16x4:  lanes 0-15 -> row=lane, VGPR{0,1}=K{0,1};
//            lanes 16-31 -> row=lane-16, VGPR{0,1}=K{2,3}
//   B 4x16:  VGPR0: lanes0-15 K=0 / lanes16-31 K=2; VGPR1: K=1 / K=3
//   C 16x16: VGPR r: lanes0-15 M=r, lanes16-31 M=r+8; N = lane&15
// --------------------------------------------------------------------------
__global__ void wmma_gemm16_kernel(const float* __restrict__ A,
                                   const float* __restrict__ W,
                                   const float* __restrict__ bias,
                                   float* __restrict__ out,
                                   int K, int N) {
    const int n0   = blockIdx.x * 16;
    const int lane = threadIdx.x;      // 0..31 (wave32)
    const int half = lane >> 4;        // 0 or 1
    const int l15  = lane & 15;

    v8f c = {};
    for (int k0 = 0; k0 < K; k0 += 4) {
        v2f a, b;
        const int ka = k0 + 2 * half;
        a.x = A[(size_t)l15 * K + ka];
        a.y = A[(size_t)l15 * K + ka + 1];
        b.x = W[(size_t)(ka)     * N + n0 + l15];
        b.y = W[(size_t)(ka + 1) * N + n0 + l15];
        // D = A*B + C  (full fp32; emits v_wmma_f32_16x16x4_f32)
        c = __builtin_amdgcn_wmma_f32_16x16x4_f32(
                /*neg_a=*/false, a, /*neg_b=*/false, b,
                /*c_mod=*/(short)0, c, /*reuse_a=*/false, /*reuse_b=*/false);
    }

    const float bn = bias[n0 + l15];
#pragma unroll
    for (int r = 0; r < 8; ++r) {
        const int m = r + 8 * half;
        out[(size_t)m * N + n0 + l15] = c[r] + bn;
    }
}

// --------------------------------------------------------------------------
// Kernel 2: bulk KV append copy.  keys[:, :4095] = k_cache, keys[:, 4095] = 0
// (same for values).  Grid-stride over float4s -> global_load/store_b128.
// --------------------------------------------------------------------------
__global__ void kv_copy_kernel(const float* __restrict__ kc,
                               const float* __restrict__ vc,
                               float* __restrict__ keys,
                               float* __restrict__ values) {
    const long per    = (long)B_DIM * S_DIM * (E_DIM / 4);   // 33,554,432 float4
    const long total  = 2 * per;
    const long stride = (long)gridDim.x * blockDim.x;
    for (long i = (long)blockIdx.x * blockDim.x + threadIdx.x; i < total; i += stride) {
        const bool  isv = (i >= per);
        const long  j   = isv ? (i - per) : i;
        const float* src = isv ? vc : kc;
        float*       dst = isv ? values : keys;
        const long e4 = j & 511;          // E/4 = 512
        const long s  = (j >> 9) & 4095;  // S = 4096
        const long b  = j >> 21;          // / (512*4096)
        float4 val;
        if (s < S_OLD) {
            val = *(const float4*)(src + ((b * S_OLD + s) * (long)E_DIM + e4 * 4));
        } else {
            val = make_float4(0.f, 0.f, 0.f, 0.f);
        }
        *(float4*)(dst + j * 4) = val;
    }
}

// --------------------------------------------------------------------------
// Kernel 3: write k_new / v_new at row last[b]
// --------------------------------------------------------------------------
__global__ void kv_append_kernel(const float* __restrict__ qkv,
                                 const int* __restrict__ last,
                                 float* __restrict__ keys,
                                 float* __restrict__ values) {
    const int b = blockIdx.x;
    const int l = last[b];
    const size_t dst = ((size_t)b * S_DIM + l) * E_DIM;
    const size_t src = (size_t)b * QKV_N;
    for (int e = threadIdx.x; e < E_DIM; e += blockDim.x) {
        keys[dst + e]   = qkv[src + E_DIM + e];      // k_new
        values[dst + e] = qkv[src + 2 * E_DIM + e];  // v_new
    }
}

// --------------------------------------------------------------------------
// Kernel 4: fused decode attention.  One block per (b,h): 256 threads = 8 waves.
//   pass1: wave w computes scores for rows s = w, w+8, ... (coalesced 512B row
//          reads, wave32 __shfl_xor reduction), + bias + mask, into LDS.
//   softmax: block max / exp / sum in LDS.
//   pass2: 8 lane-groups accumulate P*V with coalesced float4 loads.
// --------------------------------------------------------------------------
__global__ void attn_kernel(const float* __restrict__ qkv,
                            const float* __restrict__ keys,
                            const float* __restrict__ values,
                            const int* __restrict__ mask,
                            const float* __restrict__ bias,   // attn_bias row 0
                            float* __restrict__ ctx) {
    __shared__ __align__(16) float q_s[D_DIM];
    __shared__ float sc[S_DIM];
    __shared__ float red[256];
    __shared__ float ctxred[8 * D_DIM];

    const int b   = blockIdx.x >> 4;
    const int h   = blockIdx.x & 15;
    const int tid = threadIdx.x;
    const int wave = tid >> 5;
    const int lane = tid & 31;
    const float scale = 0.08838834764831845f;   // 1/sqrt(128)

    if (tid < D_DIM)
        q_s[tid] = qkv[(size_t)b * QKV_N + h * D_DIM + tid] * scale;
    __syncthreads();

    const float* Kb = keys   + ((size_t)b * S_DIM) * E_DIM + h * D_DIM;
    const float* Vb = values + ((size_t)b * S_DIM) * E_DIM + h * D_DIM;
    const float4 q4 = *(const float4*)(q_s + lane * 4);

    // ---- pass 1: scores ----
    for (int s = wave; s < S_DIM; s += 8) {
        const float4 k4 = *(const float4*)(Kb + (size_t)s * E_DIM + lane * 4);
        float p = q4.x * k4.x + q4.y * k4.y + q4.z * k4.z + q4.w * k4.w;
#pragma unroll
        for (int off = 16; off >= 1; off >>= 1)
            p += __shfl_xor(p, off, 32);
        if (lane == 0) {
            float sv = p + bias[s];
            if (mask[(size_t)b * S_DIM + s] == 0) sv = -1.0e9f;
            sc[s] = sv;
        }
    }
    __syncthreads();

    // ---- block max ----
    float m = -INFINITY;
    for (int s = tid; s < S_DIM; s += 256) m = fmaxf(m, sc[s]);
    red[tid] = m;
    __syncthreads();
    for (int w = 128; w >= 1; w >>= 1) {
        if (tid < w) red[tid] = fmaxf(red[tid], red[tid + w]);
        __syncthreads();
    }
    m = red[0];
    __syncthreads();

    // ---- exp + block sum ----
    float ssum = 0.f;
    for (int s = tid; s < S_DIM; s += 256) {
        const float p = __expf(sc[s] - m);
        sc[s] = p;
        ssum += p;
    }
    red[tid] = ssum;
    __syncthreads();
    for (int w = 128; w >= 1; w >>= 1) {
        if (tid < w) red[tid] += red[tid + w];
        __syncthreads();
    }
    const float inv = 1.0f / red[0];
    __syncthreads();

    // ---- pass 2: ctx = P * V ----
    float4 acc = make_float4(0.f, 0.f, 0.f, 0.f);
    for (int s = wave; s < S_DIM; s += 8) {
        const float  p  = sc[s];
        const float4 v4 = *(const float4*)(Vb + (size_t)s * E_DIM + lane * 4);
        acc.x += p * v4.x;  acc.y += p * v4.y;
        acc.z += p * v4.z;  acc.w += p * v4.w;
    }
    float* cr = ctxred + wave * D_DIM + lane * 4;
    cr[0] = acc.x; cr[1] = acc.y; cr[2] = acc.z; cr[3] = acc.w;
    __syncthreads();
    if (tid < D_DIM) {
        float s0 = 0.f;
#pragma unroll
        for (int g = 0; g < 8; ++g) s0 += ctxred[g * D_DIM + tid];
        ctx[(size_t)b * E_DIM + h * D_DIM + tid] = s0 * inv;
    }
}

// --------------------------------------------------------------------------
extern "C" void kernel_launch(void* const* d_in, const int* in_sizes, int n_in,
                              void* d_out, int out_size, void* d_ws, size_t ws_size,
                              hipStream_t stream) {
    const float* hs        = (const float*)d_in[0];  // (16,1,2048)
    const float* k_cache   = (const float*)d_in[1];  // (16,4095,2048)
    const float* v_cache   = (const float*)d_in[2];  // (16,4095,2048)
    const int*   mask      = (const int*)  d_in[3];  // (16,4096)
    const float* attn_bias = (const float*)d_in[4];  // (4096,4096), row 0 used
    const float* c_attn_w  = (const float*)d_in[5];  // (2048,6144)
    const float* c_attn_b  = (const float*)d_in[6];  // (6144)
    const float* c_proj_w  = (const float*)d_in[7];  // (2048,2048)
    const float* c_proj_b  = (const float*)d_in[8];  // (2048)

    // d_out = [ out | keys | values ] flat
    float* out    = (float*)d_out;
    float* keys   = out + (size_t)B_DIM * E_DIM;
    float* values = keys + (size_t)B_DIM * S_DIM * E_DIM;

    // workspace: qkv (16x6144) | ctx (16x2048) | last (16 ints)
    float* ws   = (float*)d_ws;
    float* qkv  = ws;                       //  98304 floats
    float* ctx  = ws + 98304;               //  32768 floats
    int*   last = (int*)(ws + 131072);      //  16 ints  (~513 KB total)

    // 0) last index per batch
    last_kernel<<<16, 256, 0, stream>>>(mask, last);

    // 1) QKV projection: 384 N-tiles, one wave each (v_wmma_f32_16x16x4_f32)
    wmma_gemm16_kernel<<<QKV_N / 16, 32, 0, stream>>>(hs, c_attn_w, c_attn_b,
                                                      qkv, E_DIM, QKV_N);

    // 2) bulk cache copy (~2.1 GB of traffic -> dominates; float4 streaming)
    kv_copy_kernel<<<8192, 256, 0, stream>>>(k_cache, v_cache, keys, values);

    // 3) place k_new/v_new at row last[b]
    kv_append_kernel<<<B_DIM, 256, 0, stream>>>(qkv, last, keys, values);

    // 4) fused attention, one block per (b,h)
    attn_kernel<<<B_DIM * H_DIM, 256, 0, stream>>>(qkv, keys, values, mask,
                                                   attn_bias, ctx);

    // 5) output projection
    wmma_gemm16_kernel<<<E_DIM / 16, 32, 0, stream>>>(ctx, c_proj_w, c_proj_b,
                                                      out, E_DIM, E_DIM);
}